// GCNLayer_20134806684030
// MI455X (gfx1250) — compile-verified
//
#include <hip/hip_runtime.h>
#include <hip/hip_bf16.h>
#include <stddef.h>
#include <stdint.h>

// ---------------------------------------------------------------------------
// GCN layer for MI455X (gfx1250, wave32)
//   h   = x @ W^T + b                  (fp32 WMMA 16x16x4)
//   deg = scatter-add over sym edges + self loop
//   out = relu( scatter(norm * h) )    (L2-resident f32 atomics)
// ---------------------------------------------------------------------------

typedef float v2f __attribute__((ext_vector_type(2)));
typedef float v8f __attribute__((ext_vector_type(8)));

#define DIM 128

// K0: zero the output accumulator, init deg = 1.0 (self loop)
__global__ void gcn_init_kernel(float* __restrict__ out, float* __restrict__ deg,
                                int nOut, int nNodes) {
    int i = blockIdx.x * blockDim.x + threadIdx.x;
    if (i < nOut)   out[i] = 0.0f;
    if (i < nNodes) deg[i] = 1.0f;
}

// K1: degree accumulation over undirected (symmetrized) edges
__global__ void gcn_degree_kernel(const int* __restrict__ row, const int* __restrict__ col,
                                  float* __restrict__ deg, int E) {
    int e = blockIdx.x * blockDim.x + threadIdx.x;
    if (e < E) {
        atomicAdd(&deg[row[e]], 1.0f);
        atomicAdd(&deg[col[e]], 1.0f);
    }
}

// K2: deg -> deg^{-1/2} in place (deg >= 1 always, self loop)
__global__ void gcn_rsqrt_kernel(float* __restrict__ deg, int nNodes) {
    int i = blockIdx.x * blockDim.x + threadIdx.x;
    if (i < nNodes) deg[i] = rsqrtf(deg[i]);
}

// K3: h = x @ W^T + b using V_WMMA_F32_16X16X4_F32.
// Block = 160 threads = 5 waves; each wave produces a 16-row x 128-col strip,
// so a block covers 80 rows and 1250 blocks cover N=100000 exactly (no tail).
// W ([out,in] = [128,128]) is staged in LDS via async global->LDS copies with
// a rotate swizzle so the stride-128 B-fragment reads are bank-conflict free:
//   W[o][k] stored at Wl[o*128 + ((k + 4*o) & 127)]   (bank = (k+4o) mod 64)
__global__ void __launch_bounds__(160)
gcn_gemm_wmma_kernel(const float* __restrict__ x, const float* __restrict__ W,
                     const float* __restrict__ bias, float* __restrict__ h, int nNodes) {
    __shared__ float Wl[DIM * DIM];  // 64 KB of the 320 KB WGP LDS

    // ---- Stage W into LDS with the swizzle, using GLOBAL_LOAD_ASYNC_TO_LDS_B128.
    // vdst VGPR carries the per-lane LDS byte offset (relative to the LDS base),
    // voff is the 32-bit byte offset against the SGPR base (GVS mode).
    {
        const unsigned ldsBase = (unsigned)(uintptr_t)&Wl[0];  // low 32 bits = LDS offset
        for (int i = threadIdx.x; i < DIM * DIM / 4; i += blockDim.x) {
            int o  = i >> 5;                          // output row (0..127)
            int kq = i & 31;                          // float4 index within row
            int kk = ((kq << 2) + 4 * o) & (DIM - 1); // swizzled k, 4-aligned
            unsigned dstByte = ldsBase + (unsigned)(o * DIM + kk) * 4u;
            unsigned srcByte = (unsigned)i * 16u;
            asm volatile("global_load_async_to_lds_b128 %0, %1, %2"
                         :
                         : "v"(dstByte), "v"(srcByte), "s"(W)
                         : "memory");
        }
        asm volatile("s_wait_asynccnt 0x0" ::: "memory");
    }
    __syncthreads();

    const int wave = threadIdx.x >> 5;
    const int lane = threadIdx.x & 31;
    const int nl   = lane & 15;       // M (for A) / N (for B,C,D) within tile
    const int hi   = lane >> 4;       // lane half selects K pair / row half

    const int rowBase = blockIdx.x * 80 + wave * 16;

    v8f acc[8];
#pragma unroll
    for (int nb = 0; nb < 8; ++nb) acc[nb] = (v8f)0.0f;

    // A-fragment source row (clamped for a ragged tail; with the exact launch
    // config every wave is full, and clamping keeps EXEC all-1s for WMMA).
    int aRow = rowBase + nl;
    if (aRow >= nNodes) aRow = nNodes - 1;
    const float* xrow = x + (size_t)aRow * DIM;

    for (int k0 = 0; k0 < DIM; k0 += 4) {
        const int ka = k0 + 2 * hi;          // this lane-half holds K = ka, ka+1
        v2f a;
        a.x = xrow[ka];
        a.y = xrow[ka + 1];
#pragma unroll
        for (int nb = 0; nb < 8; ++nb) {
            const int ocol = nb * 16 + nl;   // B column index (output feature)
            const int base = ocol * DIM + ((ka + 4 * ocol) & (DIM - 1));
            v2f b;
            b.x = Wl[base];
            b.y = Wl[base + 1];
            acc[nb] = __builtin_amdgcn_wmma_f32_16x16x4_f32(
                false, a, false, b, (short)0, acc[nb], false, false);
        }
    }

    // Store D (+ bias). C/D layout: lane -> N = lane%16; vgpr j -> M = j + 8*hi.
    // Uniform-per-wave full-tile test keeps the fast path branch-free.
    if (rowBase + 16 <= nNodes) {
#pragma unroll
        for (int nb = 0; nb < 8; ++nb) {
            const int ocol = nb * 16 + nl;
            const float bv = bias[ocol];
            float* hp = h + (size_t)(rowBase + 8 * hi) * DIM + ocol;
#pragma unroll
            for (int j = 0; j < 8; ++j) {
                hp[(size_t)j * DIM] = acc[nb][j] + bv;
            }
        }
    } else {
#pragma unroll
        for (int nb = 0; nb < 8; ++nb) {
            const int ocol = nb * 16 + nl;
            const float bv = bias[ocol];
#pragma unroll
            for (int j = 0; j < 8; ++j) {
                const int r = rowBase + j + 8 * hi;
                if (r < nNodes) h[(size_t)r * DIM + ocol] = acc[nb][j] + bv;
            }
        }
    }
}

// K4: one wave32 per edge; lane owns a float4 slice of the 128-dim feature.
// Handles both directions (u->v and v->u) so dinv/h rows are reused from L2.
__global__ void __launch_bounds__(256)
gcn_scatter_kernel(const int* __restrict__ row, const int* __restrict__ col,
                   const float* __restrict__ dinv, const float* __restrict__ h,
                   float* __restrict__ out, int E) {
    const int gtid = blockIdx.x * blockDim.x + threadIdx.x;
    const int edge = gtid >> 5;
    const int lane = gtid & 31;
    if (edge >= E) return;  // uniform per wave

    const int u = row[edge];
    const int v = col[edge];
    const float n = dinv[u] * dinv[v];

    const float4 hu = ((const float4*)(h + (size_t)u * DIM))[lane];
    const float4 hv = ((const float4*)(h + (size_t)v * DIM))[lane];

    float* ov = out + (size_t)v * DIM + lane * 4;
    float* ou = out + (size_t)u * DIM + lane * 4;

    atomicAdd(ov + 0, n * hu.x);
    atomicAdd(ov + 1, n * hu.y);
    atomicAdd(ov + 2, n * hu.z);
    atomicAdd(ov + 3, n * hu.w);

    atomicAdd(ou + 0, n * hv.x);
    atomicAdd(ou + 1, n * hv.y);
    atomicAdd(ou + 2, n * hv.z);
    atomicAdd(ou + 3, n * hv.w);
}

// K5: out = relu(out + dinv^2 * h)   (self-loop term folded in here)
__global__ void gcn_finalize_kernel(const float* __restrict__ h, const float* __restrict__ dinv,
                                    float* __restrict__ out, int nNodes) {
    const int i = blockIdx.x * blockDim.x + threadIdx.x;
    if (i >= nNodes * 32) return;
    const int node = i >> 5;
    const int q    = i & 31;
    float s = dinv[node];
    s = s * s;
    const float4 hv = ((const float4*)(h + (size_t)node * DIM))[q];
    float4* op = (float4*)(out + (size_t)node * DIM) + q;
    float4 o = *op;
    o.x = fmaxf(o.x + s * hv.x, 0.0f);
    o.y = fmaxf(o.y + s * hv.y, 0.0f);
    o.z = fmaxf(o.z + s * hv.z, 0.0f);
    o.w = fmaxf(o.w + s * hv.w, 0.0f);
    *op = o;
}

extern "C" void kernel_launch(void* const* d_in, const int* in_sizes, int n_in,
                              void* d_out, int out_size, void* d_ws, size_t ws_size,
                              hipStream_t stream) {
    const float* x    = (const float*)d_in[0];
    const int*   ei   = (const int*)d_in[1];
    const float* W    = (const float*)d_in[2];
    const float* bias = (const float*)d_in[3];

    const int nNodes = in_sizes[0] / DIM;   // 100000
    const int E      = in_sizes[1] / 2;     // 800000
    const int*   row = ei;
    const int*   col = ei + E;

    float* out = (float*)d_out;

    // Workspace layout: [ h : nNodes*128 f32 | deg/dinv : nNodes f32 ]
    float* h    = (float*)d_ws;
    float* deg  = (float*)((char*)d_ws + (size_t)nNodes * DIM * sizeof(float));

    const int nOut = nNodes * DIM;

    // K0: zero out-accumulator, deg = 1
    {
        int threads = 256, blocks = (nOut + threads - 1) / threads;
        gcn_init_kernel<<<blocks, threads, 0, stream>>>(out, deg, nOut, nNodes);
    }
    // K1: degrees
    {
        int threads = 256, blocks = (E + threads - 1) / threads;
        gcn_degree_kernel<<<blocks, threads, 0, stream>>>(row, col, deg, E);
    }
    // K2: dinv
    {
        int threads = 256, blocks = (nNodes + threads - 1) / threads;
        gcn_rsqrt_kernel<<<blocks, threads, 0, stream>>>(deg, nNodes);
    }
    // K3: h = x W^T + b  (WMMA, 5 waves/block -> 80 rows/block, no tail at N=100000)
    {
        int threads = 160;
        int blocks  = (nNodes + 79) / 80;
        gcn_gemm_wmma_kernel<<<blocks, threads, 0, stream>>>(x, W, bias, h, nNodes);
    }
    // K4: edge scatter (one wave per edge)
    {
        long long totalThreads = (long long)E * 32;
        int threads = 256;
        int blocks  = (int)((totalThreads + threads - 1) / threads);
        gcn_scatter_kernel<<<blocks, threads, 0, stream>>>(row, col, deg, h, out, E);
    }
    // K5: finalize (self loop + relu)
    {
        long long totalThreads = (long long)nNodes * 32;
        int threads = 256;
        int blocks  = (int)((totalThreads + threads - 1) / threads);
        gcn_finalize_kernel<<<blocks, threads, 0, stream>>>(h, deg, out, nNodes);
    }
}